// GlobalPointer_31293131719279
// MI455X (gfx1250) — compile-verified
//
#include <hip/hip_runtime.h>
#include <hip/hip_bf16.h>

#define NUM_HEAD 12
#define HEAD_SIZE 64

// Problem constants (fixed by the reference)
constexpr int Bsz  = 8;
constexpr int S    = 1024;
constexpr int IN   = 768;
constexpr int OUTC = NUM_HEAD * HEAD_SIZE * 2;   // 1536
constexpr int M1   = Bsz * S;                    // 8192 rows for phase-1 GEMM
constexpr int KCH  = IN / 16;                    // 48 K-chunks of 16

typedef float v2f __attribute__((ext_vector_type(2)));
typedef float v8f __attribute__((ext_vector_type(8)));

#if defined(__has_builtin)
#if __has_builtin(__builtin_amdgcn_tensor_load_to_lds) && \
    __has_builtin(__builtin_amdgcn_s_wait_tensorcnt)
#define USE_TDM 1
#endif
#endif
#ifndef USE_TDM
#define USE_TDM 0
#endif

#if USE_TDM
typedef unsigned int v4u __attribute__((ext_vector_type(4)));
typedef int          v8i __attribute__((ext_vector_type(8)));
typedef int          v4i __attribute__((ext_vector_type(4)));

// Issue one TDM 2D tile load (global -> LDS) with hardware LDS padding.
// Descriptor packing per CDNA5 ISA (D# groups 0..3).
__device__ __forceinline__ void tdm_load_2d(unsigned int lds_byte, const void* gptr,
                                            unsigned int tile_w, unsigned int tile_h,
                                            unsigned long long row_stride,
                                            unsigned int pad_interval_code,
                                            unsigned int pad_amount_code)
{
    unsigned long long ga = (unsigned long long)(uintptr_t)gptr;

    v4u g0 = {
        1u,                                               // count=1, user D#
        lds_byte,                                         // lds_addr
        (unsigned int)ga,                                 // global_addr[31:0]
        ((unsigned int)(ga >> 32) & 0x01FFFFFFu) | (2u << 30) // ga[56:32] | type=2
    };

    unsigned int w0 = (2u << 16)                 // data_size = 4 bytes
                    | (1u << 20)                 // pad_enable
                    | (pad_interval_code << 22)
                    | (pad_amount_code << 25);
    unsigned int td0 = tile_w;
    unsigned int td1 = tile_h;
    unsigned long long s0 = row_stride;
    v8i g1 = {
        (int)w0,
        (int)((td0 & 0xFFFFu) << 16),                        // abar=0 | td0.lo
        (int)((td0 >> 16) | ((td1 & 0xFFFFu) << 16)),        // td0.hi | td1.lo
        (int)((td1 >> 16) | (tile_w << 16)),                 // td1.hi | tile_dim0
        (int)(tile_h & 0xFFFFu),                             // tile_dim1 | tile_dim2=0
        (int)(unsigned int)(s0 & 0xFFFFFFFFu),               // stride0.lo
        (int)(unsigned int)((s0 >> 32) & 0xFFFFu),           // stride0.hi | stride1.lo=0
        0                                                    // stride1.hi
    };

    v4i g2 = { 1, 1, 0, 0 };            // tensor_dim2=1, tensor_dim3=1
    v4i g3 = { 0, 1 << 16, 0, 0 };      // tensor_dim4=1

    v8i g4 = { 0, 0, 0, 0, 0, 0, 0, 0 };  // extra operand on clang-23 builtin

    __builtin_amdgcn_tensor_load_to_lds(g0, g1, g2, g3, g4, 0);
}
#endif  // USE_TDM

// D = A(16x4) * B(4x16) + C, fp32 WMMA (wave32).
__device__ __forceinline__ v8f wmma4(v2f a, v2f b, v8f c) {
    return __builtin_amdgcn_wmma_f32_16x16x4_f32(false, a, false, b, (short)0, c,
                                                 false, false);
}

// 16x4 fragment load: lanes 0-15 hold K={k4,k4+1} of row (row0+lane),
// lanes 16-31 hold K={k4+2,k4+3}.
__device__ __forceinline__ v2f frag_ld(const float* s, int row0, int ldr, int k4) {
    const int lane = threadIdx.x & 31;
    const float* p = s + (row0 + (lane & 15)) * ldr + k4 + ((lane >> 4) << 1);
    return *(const v2f*)p;
}

// ---------------------------------------------------------------------------
// Kernel 1: hidden = X@W + b, fused RoPE, split into q_ws / k_ws [b][h][s][d]
// grid (64, 12), block 128.  Double-buffered LDS; TDM DMA overlaps compute.
// Each block covers one head (128 N-channels) and 128 consecutive positions,
// so a 128x32 sin/cos LDS table serves the whole RoPE epilogue.
// ---------------------------------------------------------------------------
__global__ void __launch_bounds__(128)
gemm_rope_kernel(const float* __restrict__ X, const float* __restrict__ W,
                 const float* __restrict__ bias,
                 float* __restrict__ qws, float* __restrict__ kws)
{
    __shared__ float  As[2][128 * 18];   // [m][k], pitch 18 (TDM pads 16+2)
    __shared__ float  Bs[2][128 * 18];   // [n][k] (transposed W tile)
    __shared__ float2 SC[128 * 32];      // [pos_local][freq] -> (sin, cos)

    const int t    = threadIdx.x;
    const int mblk = blockIdx.x;     // 0..63
    const int nblk = blockIdx.y;     // 0..11 == head index
    const int wave = t >> 5;
    const int wm   = (wave >> 1) * 64;
    const int wn   = (wave & 1)  * 64;

    // ---- sin/cos table: 32 entries per thread, once per block ----
#pragma unroll
    for (int j = 0; j < 32; ++j) {
        const int e  = t * 32 + j;           // 0..4095
        const int pl = e >> 5;               // pos_local 0..127
        const int fi = e & 31;               // freq index = d/2
        const int pos = (mblk * 128 + pl) & (S - 1);
        const float invf = __powf(10000.0f, -(float)fi * (1.0f / 32.0f));
        float sn, cs;
        __sincosf((float)pos * invf, &sn, &cs);
        SC[e] = make_float2(sn, cs);
    }

    const v8f vzero = {0.f, 0.f, 0.f, 0.f, 0.f, 0.f, 0.f, 0.f};
    v8f acc[4][4];
#pragma unroll
    for (int mm = 0; mm < 4; ++mm)
#pragma unroll
        for (int nn = 0; nn < 4; ++nn) acc[mm][nn] = vzero;

#if USE_TDM
    const float* xtile0 = X + (size_t)(mblk * 128) * IN;
#else
    const float* xrow = X + (size_t)(mblk * 128 + t) * IN;
    float xv[16];
#endif
    // W tile: thread (krow, cg) loads 16 floats of one W row, scatters transposed.
    const int krow = t >> 3;        // 0..15
    const int cg   = t & 7;         // 0..7
    const float* wrow = W + (size_t)krow * OUTC + nblk * 128 + cg * 16;
    float wv[16];

    // ---- prologue: stage chunk 0 into buffer 0 ----
#if USE_TDM
    if (wave == 0)
        tdm_load_2d((unsigned int)(uintptr_t)(void*)As[0], xtile0, 16, 128, IN, 3, 1);
#else
#pragma unroll
    for (int j = 0; j < 4; ++j) {
        float4 v = ((const float4*)xrow)[j];
        xv[j * 4 + 0] = v.x; xv[j * 4 + 1] = v.y;
        xv[j * 4 + 2] = v.z; xv[j * 4 + 3] = v.w;
    }
#pragma unroll
    for (int j = 0; j < 8; ++j)
        ((float2*)&As[0][t * 18])[j] = make_float2(xv[2 * j], xv[2 * j + 1]);
#endif
#pragma unroll
    for (int j = 0; j < 4; ++j) {
        float4 v = ((const float4*)wrow)[j];
        wv[j * 4 + 0] = v.x; wv[j * 4 + 1] = v.y;
        wv[j * 4 + 2] = v.z; wv[j * 4 + 3] = v.w;
    }
#pragma unroll
    for (int j = 0; j < 16; ++j)
        Bs[0][(cg * 16 + j) * 18 + krow] = wv[j];
#if USE_TDM
    if (wave == 0) __builtin_amdgcn_s_wait_tensorcnt(0);
#endif
    __syncthreads();

    // ---- main loop: compute chunk kc while DMA'ing chunk kc+1 ----
    for (int kc = 0; kc < KCH; ++kc) {
        const int cur = kc & 1;
        const int nxt = cur ^ 1;
        const bool more = (kc + 1) < KCH;

        if (more) {
#if USE_TDM
            if (wave == 0)
                tdm_load_2d((unsigned int)(uintptr_t)(void*)As[nxt],
                            xtile0 + (kc + 1) * 16, 16, 128, IN, 3, 1);
#else
#pragma unroll
            for (int j = 0; j < 4; ++j) {
                float4 v = ((const float4*)(xrow + (kc + 1) * 16))[j];
                xv[j * 4 + 0] = v.x; xv[j * 4 + 1] = v.y;
                xv[j * 4 + 2] = v.z; xv[j * 4 + 3] = v.w;
            }
#endif
#pragma unroll
            for (int j = 0; j < 4; ++j) {
                float4 v = ((const float4*)(wrow + (size_t)(kc + 1) * 16 * OUTC))[j];
                wv[j * 4 + 0] = v.x; wv[j * 4 + 1] = v.y;
                wv[j * 4 + 2] = v.z; wv[j * 4 + 3] = v.w;
            }
        }

#pragma unroll
        for (int k4 = 0; k4 < 16; k4 += 4) {
            v2f a[4], b[4];
#pragma unroll
            for (int mm = 0; mm < 4; ++mm)
                a[mm] = frag_ld(As[cur], wm + mm * 16, 18, k4);
#pragma unroll
            for (int nn = 0; nn < 4; ++nn)
                b[nn] = frag_ld(Bs[cur], wn + nn * 16, 18, k4);
#pragma unroll
            for (int mm = 0; mm < 4; ++mm)
#pragma unroll
                for (int nn = 0; nn < 4; ++nn)
                    acc[mm][nn] = wmma4(a[mm], b[nn], acc[mm][nn]);
        }

        if (more) {
#if !USE_TDM
#pragma unroll
            for (int j = 0; j < 8; ++j)
                ((float2*)&As[nxt][t * 18])[j] = make_float2(xv[2 * j], xv[2 * j + 1]);
#endif
#pragma unroll
            for (int j = 0; j < 16; ++j)
                Bs[nxt][(cg * 16 + j) * 18 + krow] = wv[j];
        }
#if USE_TDM
        if (wave == 0) __builtin_amdgcn_s_wait_tensorcnt(0);
#endif
        __syncthreads();
    }

    // ---- epilogue: bias + RoPE (partner via shfl_xor, sin/cos from LDS) ----
    const int lane = t & 31;
    const int half = lane >> 4;
    const int col  = lane & 15;
    const bool isq = (wn == 0);
    float* dbase = isq ? qws : kws;

#pragma unroll
    for (int mm = 0; mm < 4; ++mm)
#pragma unroll
        for (int nn = 0; nn < 4; ++nn) {
            v8f cf = acc[mm][nn];
            const int d = nn * 16 + col;          // 0..63 within head half
            const float b = bias[nblk * 128 + wn + d];
#pragma unroll
            for (int r = 0; r < 8; ++r) {
                const int pl  = wm + mm * 16 + r + half * 8;   // 0..127
                const int m   = mblk * 128 + pl;
                const int pos = m & (S - 1);
                const int bi  = m >> 10;

                float v = cf[r] + b;
                float partner = __shfl_xor(v, 1, 32);
                float2 sc = SC[pl * 32 + (d >> 1)];
                const float y = (d & 1) ? (v * sc.y + partner * sc.x)
                                        : (v * sc.y - partner * sc.x);

                const size_t dst =
                    (((size_t)(bi * NUM_HEAD + nblk) * S + pos) << 6) + d;
                dbase[dst] = y;
            }
        }
}

// ---------------------------------------------------------------------------
// Kernel 2: logits[bh] = (Q[bh] @ K[bh]^T) / sqrt(12)
// grid (8, 8, 96), block 128 (4 waves, 2x2 of 64x64 -> 128x128 tile)
// ---------------------------------------------------------------------------
__global__ void __launch_bounds__(128)
qk_kernel(const float* __restrict__ qws, const float* __restrict__ kws,
          float* __restrict__ out)
{
    __shared__ float Qs[128 * 66];   // [m][d], pitch 66 (TDM pads 64+2)
    __shared__ float Ks[128 * 66];   // [n][d], pitch 66

    const int t    = threadIdx.x;
    const int nblk = blockIdx.x;
    const int mblk = blockIdx.y;
    const int bh   = blockIdx.z;
    const int wave = t >> 5;

    const float* Qg = qws + (size_t)bh * S * HEAD_SIZE
                          + (size_t)(mblk * 128) * HEAD_SIZE;
    const float* Kg = kws + (size_t)bh * S * HEAD_SIZE
                          + (size_t)(nblk * 128) * HEAD_SIZE;

#if USE_TDM
    if (wave == 0) {
        tdm_load_2d((unsigned int)(uintptr_t)(void*)Qs, Qg, 64, 128, HEAD_SIZE, 5, 1);
        tdm_load_2d((unsigned int)(uintptr_t)(void*)Ks, Kg, 64, 128, HEAD_SIZE, 5, 1);
        __builtin_amdgcn_s_wait_tensorcnt(0);
    }
#else
    const float* Qb = Qg + (size_t)t * HEAD_SIZE;
    const float* Kb = Kg + (size_t)t * HEAD_SIZE;
#pragma unroll
    for (int j = 0; j < 16; ++j) {
        float4 q4 = ((const float4*)Qb)[j];
        float4 k4 = ((const float4*)Kb)[j];
        float2* qd = (float2*)&Qs[t * 66 + j * 4];
        float2* kd = (float2*)&Ks[t * 66 + j * 4];
        qd[0] = make_float2(q4.x, q4.y);
        qd[1] = make_float2(q4.z, q4.w);
        kd[0] = make_float2(k4.x, k4.y);
        kd[1] = make_float2(k4.z, k4.w);
    }
#endif
    __syncthreads();

    const int wm = (wave >> 1) * 64;
    const int wn = (wave & 1)  * 64;

    const v8f vzero = {0.f, 0.f, 0.f, 0.f, 0.f, 0.f, 0.f, 0.f};
    v8f acc[4][4];
#pragma unroll
    for (int mm = 0; mm < 4; ++mm)
#pragma unroll
        for (int nn = 0; nn < 4; ++nn) acc[mm][nn] = vzero;

#pragma unroll
    for (int k4 = 0; k4 < HEAD_SIZE; k4 += 4) {
        v2f a[4], b[4];
#pragma unroll
        for (int mm = 0; mm < 4; ++mm) a[mm] = frag_ld(Qs, wm + mm * 16, 66, k4);
#pragma unroll
        for (int nn = 0; nn < 4; ++nn) b[nn] = frag_ld(Ks, wn + nn * 16, 66, k4);
#pragma unroll
        for (int mm = 0; mm < 4; ++mm)
#pragma unroll
            for (int nn = 0; nn < 4; ++nn)
                acc[mm][nn] = wmma4(a[mm], b[nn], acc[mm][nn]);
    }

    const float scale = 0.28867513459481287f;    // 1/sqrt(12)
    const int lane = t & 31;
    const int half = lane >> 4;
    const int col  = lane & 15;
    const int m0   = mblk * 128 + wm;
    const int n0   = nblk * 128 + wn;
    float* obase = out + (size_t)bh * S * S;

#pragma unroll
    for (int mm = 0; mm < 4; ++mm)
#pragma unroll
        for (int nn = 0; nn < 4; ++nn) {
            v8f cf = acc[mm][nn];
#pragma unroll
            for (int r = 0; r < 8; ++r) {
                const int m = m0 + mm * 16 + r + half * 8;
                const int n = n0 + nn * 16 + col;
                obase[(size_t)m * S + n] = cf[r] * scale;
            }
        }
}

extern "C" void kernel_launch(void* const* d_in, const int* in_sizes, int n_in,
                              void* d_out, int out_size, void* d_ws, size_t ws_size,
                              hipStream_t stream) {
    const float* X    = (const float*)d_in[0];   // (8,1024,768)
    const float* W    = (const float*)d_in[1];   // (768,1536)
    const float* bias = (const float*)d_in[2];   // (1536,)
    float* out = (float*)d_out;                  // (8,12,1024,1024)

    float* qws = (float*)d_ws;                                   // 8*12*1024*64
    float* kws = qws + (size_t)Bsz * NUM_HEAD * S * HEAD_SIZE;   // same size

    dim3 g1(M1 / 128, OUTC / 128);   // (64, 12)
    gemm_rope_kernel<<<g1, dim3(128), 0, stream>>>(X, W, bias, qws, kws);

    dim3 g2(S / 128, S / 128, Bsz * NUM_HEAD);   // (8, 8, 96)
    qk_kernel<<<g2, dim3(128), 0, stream>>>(qws, kws, out);
}